// FM_3393024163983
// MI455X (gfx1250) — compile-verified
//
#include <hip/hip_runtime.h>

#define B_TOTAL 16384
#define F_CAT 26
#define CARD 100000
#define F_NUM 13
#define D_DIM 16
#define SPB 64          // samples per block
#define LDS_STRIDE 20   // floats; 80B keeps float4 (16B) alignment, breaks bank patterns

typedef __attribute__((ext_vector_type(2))) float v2f;
typedef __attribute__((ext_vector_type(8))) float v8f;

__global__ __launch_bounds__(256) void fm_fwd_kernel(
    const int*   __restrict__ catX,   // [B, 26]
    const float* __restrict__ numX,   // [B, 13]
    const float* __restrict__ Wc,     // [26, 100000, 16]
    const float* __restrict__ bc,     // [26, 100000, 1]
    const float* __restrict__ Wn,     // [13, 16]
    const float* __restrict__ bn,     // [13, 1]
    const float* __restrict__ b0,     // [1]
    float*       __restrict__ out)    // [B, 1]
{
    __shared__ float s_num[SPB * LDS_STRIDE];

    const int t = threadIdx.x;

    // ---- Phase 1 (waves 0..3): s_num = X_num(64x16, K-padded) * Wn(16x16, K-padded)
    // via v_wmma_f32_16x16x4_f32, one 16-sample tile per wave, K=16 in 4 steps.
    if (t < 128) {
        const int w = t >> 5;          // wave -> tile 0..3
        const int L = t & 31;
        const int h = L >> 4;          // lane half
        const int m = L & 15;          // A: row M / B: col N
        const int row = blockIdx.x * SPB + w * 16 + m;   // global sample for A
        v8f acc = {0.f, 0.f, 0.f, 0.f, 0.f, 0.f, 0.f, 0.f};
#pragma unroll
        for (int c = 0; c < 4; ++c) {
            const int k0 = 4 * c + 2 * h;   // ISA 16x4 f32 A layout: lane half selects K pair
            const int k1 = k0 + 1;
            v2f a, bm;
            a[0]  = (k0 < F_NUM) ? numX[row * F_NUM + k0] : 0.f;
            a[1]  = (k1 < F_NUM) ? numX[row * F_NUM + k1] : 0.f;
            bm[0] = (k0 < F_NUM) ? Wn[k0 * D_DIM + m] : 0.f;
            bm[1] = (k1 < F_NUM) ? Wn[k1 * D_DIM + m] : 0.f;
            acc = __builtin_amdgcn_wmma_f32_16x16x4_f32(
                false, a, false, bm, (short)0, acc, false, false);
        }
        // D layout: vgpr r -> M = r + 8*h (sample), N = m (dim)
#pragma unroll
        for (int r = 0; r < 8; ++r)
            s_num[(w * 16 + r + 8 * h) * LDS_STRIDE + m] = acc[r];
    }
    __syncthreads();

    // ---- Phase 2 (all 256 threads): gather-bound FM body, 4 lanes per sample.
    const int g = t >> 2;                        // local sample 0..63
    const int j = t & 3;                         // quad lane: owns dims j*4..j*4+3
    const int b = blockIdx.x * SPB + g;          // global sample

    int idx[F_CAT];
#pragma unroll
    for (int f = 0; f < F_CAT; ++f) idx[f] = catX[b * F_CAT + f];

    float sx = 0.f, sy = 0.f, sz = 0.f, sw = 0.f;   // partial s (4 dims)
    float q = 0.f, bias = 0.f;

    // categorical 2nd-order gathers: 26 independent b128 loads per lane
#pragma unroll
    for (int f = 0; f < F_CAT; ++f) {
        const float4* row = reinterpret_cast<const float4*>(
            Wc + ((size_t)f * CARD + (size_t)idx[f]) * D_DIM);
        const float4 v = row[j];
        sx += v.x; sy += v.y; sz += v.z; sw += v.w;
        q  += v.x * v.x + v.y * v.y + v.z * v.z + v.w * v.w;
    }
    // categorical 1st-order biases, split across the quad
#pragma unroll
    for (int f = 0; f < 7; ++f) {
        const int ff = j + 4 * f;
        if (ff < F_CAT) bias += bc[(size_t)ff * CARD + (size_t)idx[ff]];
    }

    // numeric part: q contribution factors as x_f^2 * ||Wn[f]||^2 (per-lane 4-dim slice)
    float xv[F_NUM];
#pragma unroll
    for (int f = 0; f < F_NUM; ++f) xv[f] = numX[b * F_NUM + f];
#pragma unroll
    for (int f = 0; f < F_NUM; ++f) {
        const float4 wv = reinterpret_cast<const float4*>(Wn + f * D_DIM)[j];
        q += xv[f] * xv[f] * (wv.x * wv.x + wv.y * wv.y + wv.z * wv.z + wv.w * wv.w);
    }
#pragma unroll
    for (int f = 0; f < 4; ++f) {
        const int ff = j + 4 * f;
        if (ff < F_NUM) bias += xv[ff] * bn[ff];
    }

    // fold in the WMMA-produced numeric factor sums
    const float4 sn = *reinterpret_cast<const float4*>(&s_num[g * LDS_STRIDE + j * 4]);
    sx += sn.x; sy += sn.y; sz += sn.z; sw += sn.w;

    float p = sx * sx + sy * sy + sz * sz + sw * sw;
    // quad reductions (wave32: xor masks 1,2 stay inside the quad)
    p    += __shfl_xor(p, 1);    p    += __shfl_xor(p, 2);
    q    += __shfl_xor(q, 1);    q    += __shfl_xor(q, 2);
    bias += __shfl_xor(bias, 1); bias += __shfl_xor(bias, 2);

    if (j == 0)
        out[b] = bias + b0[0] + 0.5f * (p - q);
}

extern "C" void kernel_launch(void* const* d_in, const int* in_sizes, int n_in,
                              void* d_out, int out_size, void* d_ws, size_t ws_size,
                              hipStream_t stream) {
    (void)in_sizes; (void)n_in; (void)out_size; (void)d_ws; (void)ws_size;
    const int*   catX = (const int*)  d_in[0];
    const float* numX = (const float*)d_in[1];
    const float* Wc   = (const float*)d_in[2];
    const float* bc   = (const float*)d_in[3];
    const float* Wn   = (const float*)d_in[4];
    const float* bn   = (const float*)d_in[5];
    const float* b0   = (const float*)d_in[6];
    float* out = (float*)d_out;

    dim3 grid(B_TOTAL / SPB);   // 256 blocks
    dim3 block(256);            // 8 waves; waves 0..3 run the WMMA tile GEMM
    fm_fwd_kernel<<<grid, block, 0, stream>>>(catX, numX, Wc, bc, Wn, bn, b0, out);
}